// GCNClassifier_76347338653853
// MI455X (gfx1250) — compile-verified
//
#include <hip/hip_runtime.h>

// ---------------------------------------------------------------------------
// GCN classifier for MI455X (gfx1250, wave32).
// Pipeline: degree -> dinv -> h1=x@W1 (+selfloop init) -> edge scatter (16f)
//           -> relu(+b1) -> WMMA f32 16x16x4 GEMM h1post@W2 (+selfloop init)
//           -> edge scatter (64f, atomic f32 in L2) -> sorted-batch pooling
//           -> tiny MLP -> d_out [128,2] f32
// ---------------------------------------------------------------------------

#define N_NODES   131072
#define N_EDGES   4194304
#define NUM_GRAPHS 128

typedef float v2f_t __attribute__((ext_vector_type(2)));
typedef float v8f_t __attribute__((ext_vector_type(8)));

// ---- degree / dinv ---------------------------------------------------------

__global__ void k_init_deg(int* __restrict__ deg) {
    int v = blockIdx.x * blockDim.x + threadIdx.x;
    if (v < N_NODES) deg[v] = 1;   // self-loop contribution
}

__global__ void k_degree(const int* __restrict__ edge_dst, int* __restrict__ deg) {
    int e = blockIdx.x * blockDim.x + threadIdx.x;
    if (e < N_EDGES) atomicAdd(&deg[edge_dst[e]], 1);
}

__global__ void k_dinv(const int* __restrict__ deg, float* __restrict__ dinv) {
    int v = blockIdx.x * blockDim.x + threadIdx.x;
    if (v < N_NODES) dinv[v] = rsqrtf((float)deg[v]);
}

// ---- layer 1: h1 = x@W1, agg1 init = dinv^2 * h1 ---------------------------

__global__ void k_conv1_node(const float* __restrict__ x, const float* __restrict__ W1,
                             const float* __restrict__ dinv,
                             float* __restrict__ h1, float* __restrict__ agg1) {
    int v = blockIdx.x * blockDim.x + threadIdx.x;
    if (v >= N_NODES) return;
    float x0 = x[3 * v + 0], x1 = x[3 * v + 1], x2 = x[3 * v + 2];
    float dv = dinv[v];
    float dv2 = dv * dv;
#pragma unroll
    for (int f = 0; f < 16; ++f) {
        float h = x0 * W1[f] + x1 * W1[16 + f] + x2 * W1[32 + f];
        h1[v * 16 + f]  = h;
        agg1[v * 16 + f] = dv2 * h;   // self-loop message
    }
}

// ---- edge scatter, 16 features (4 float4 chunks per edge) ------------------

__global__ void k_scatter16(const int* __restrict__ esrc, const int* __restrict__ edst,
                            const float* __restrict__ dinv,
                            const float* __restrict__ h1, float* __restrict__ agg1) {
    int t = blockIdx.x * blockDim.x + threadIdx.x;   // N_EDGES*4 threads
    int e = t >> 2;
    int c = t & 3;
    if (e >= N_EDGES) return;
    int s = esrc[e], d = edst[e];
    float w = dinv[s] * dinv[d];
    float4 m = *reinterpret_cast<const float4*>(h1 + s * 16 + c * 4);
    float* out = agg1 + d * 16 + c * 4;
    atomicAdd(out + 0, w * m.x);
    atomicAdd(out + 1, w * m.y);
    atomicAdd(out + 2, w * m.z);
    atomicAdd(out + 3, w * m.w);
}

// ---- bias + relu in place --------------------------------------------------

__global__ void k_bias_relu16(float* __restrict__ agg1, const float* __restrict__ b1) {
    int i = blockIdx.x * blockDim.x + threadIdx.x;   // N_NODES*16 threads
    if (i < N_NODES * 16) agg1[i] = fmaxf(agg1[i] + b1[i & 15], 0.0f);
}

// ---- WMMA GEMM: hW = h1post[131072,16] @ W2[16,64]; agg2 = dinv^2 * hW -----
// One wave per 16x16 output tile; 4 x V_WMMA_F32_16X16X4_F32 over K=16.
// A (16x4 f32): lanes 0-15 hold (M=lane, K={c4+0,c4+1}), lanes 16-31 (M=lane-16, K={c4+2,c4+3}).
// B (4x16 f32): same K split across lane halves, N = lane&15.
// D (16x16 f32): VGPR r: lanes 0-15 -> M=r, lanes 16-31 -> M=8+r, N = lane&15.

__global__ void k_gemm16_64_wmma(const float* __restrict__ hpost,
                                 const float* __restrict__ W2,
                                 const float* __restrict__ dinv,
                                 float* __restrict__ hW,
                                 float* __restrict__ agg2) {
    const int wave = threadIdx.x >> 5;
    const int lane = threadIdx.x & 31;
    const int tile = blockIdx.x * 8 + wave;      // 32768 tiles total (exact)
    const int mt = tile >> 2;                    // 8192 row tiles
    const int nt = tile & 3;                     // 4 col tiles
    const int m0 = mt << 4;
    const int n0 = nt << 4;
    const int rlo = lane & 15;
    const int kh  = lane >> 4;                   // 0: K lo pair, 1: K hi pair

    v8f_t acc = {};
    const float* arow = hpost + (m0 + rlo) * 16 + (kh << 1);
#pragma unroll
    for (int c = 0; c < 4; ++c) {
        v2f_t a, b;
        a.x = arow[c * 4 + 0];
        a.y = arow[c * 4 + 1];
        const int kb = c * 4 + (kh << 1);
        b.x = W2[kb * 64 + n0 + rlo];
        b.y = W2[(kb + 1) * 64 + n0 + rlo];
        acc = __builtin_amdgcn_wmma_f32_16x16x4_f32(
            /*neg_a=*/false, a, /*neg_b=*/false, b,
            /*c_mod=*/(short)0, acc, /*reuse_a=*/false, /*reuse_b=*/false);
    }
#pragma unroll
    for (int r = 0; r < 8; ++r) {
        const int row = m0 + r + (kh << 3);
        const int col = n0 + rlo;
        const float v = acc[r];
        const float dv = dinv[row];
        hW[row * 64 + col]   = v;
        agg2[row * 64 + col] = dv * dv * v;      // self-loop message init
    }
}

// ---- edge scatter, 64 features (16 float4 chunks per edge) -----------------

__global__ void k_scatter64(const int* __restrict__ esrc, const int* __restrict__ edst,
                            const float* __restrict__ dinv,
                            const float* __restrict__ hW, float* __restrict__ agg2) {
    int t = blockIdx.x * blockDim.x + threadIdx.x;   // N_EDGES*16 threads
    int e = t >> 4;
    int c = t & 15;
    if (e >= N_EDGES) return;
    if (e + 4096 < N_EDGES) {                        // stream-ahead hint on edge list
        __builtin_prefetch(&esrc[e + 4096], 0, 0);
    }
    int s = esrc[e], d = edst[e];
    float w = dinv[s] * dinv[d];
    float4 m = *reinterpret_cast<const float4*>(hW + s * 64 + c * 4);
    float* out = agg2 + d * 64 + c * 4;
    atomicAdd(out + 0, w * m.x);
    atomicAdd(out + 1, w * m.y);
    atomicAdd(out + 2, w * m.z);
    atomicAdd(out + 3, w * m.w);
}

// ---- pooling: batch is sorted -> binary search per graph, no atomics -------

__global__ void k_pool(const float* __restrict__ agg2, const float* __restrict__ b2,
                       const int* __restrict__ batch, float* __restrict__ pooled) {
    const int g = blockIdx.x;                        // 128 blocks
    // lower_bound(batch, g) and lower_bound(batch, g+1)
    int lo = 0, hi = N_NODES;
    while (lo < hi) { int mid = (lo + hi) >> 1; if (batch[mid] < g) lo = mid + 1; else hi = mid; }
    const int start = lo;
    lo = start; hi = N_NODES;
    while (lo < hi) { int mid = (lo + hi) >> 1; if (batch[mid] < g + 1) lo = mid + 1; else hi = mid; }
    const int end = lo;

    const int f = threadIdx.x & 63;
    const int j = threadIdx.x >> 6;                  // 4 node-lanes per feature
    const float bias = b2[f];
    float local = 0.0f;
    for (int v = start + j; v < end; v += 4)
        local += fmaxf(agg2[v * 64 + f] + bias, 0.0f);

    __shared__ float red[256];
    red[threadIdx.x] = local;
    __syncthreads();
    if (j == 0) {
        float s = red[f] + red[f + 64] + red[f + 128] + red[f + 192];
        int cnt = end - start;
        pooled[g * 64 + f] = s / (float)(cnt > 0 ? cnt : 1);
    }
}

// ---- final MLP: [128,64] -> 32 -> 2 (no activation, per reference) ---------

__global__ void k_mlp(const float* __restrict__ pooled,
                      const float* __restrict__ fcw1, const float* __restrict__ fcb1,
                      const float* __restrict__ fcw2, const float* __restrict__ fcb2,
                      float* __restrict__ out) {
    int g = threadIdx.x;
    if (g >= NUM_GRAPHS) return;
    const float* p = pooled + g * 64;
    float o0 = fcb2[0], o1 = fcb2[1];
    for (int i = 0; i < 32; ++i) {
        float h = fcb1[i];
#pragma unroll 8
        for (int k = 0; k < 64; ++k) h += p[k] * fcw1[k * 32 + i];
        o0 += h * fcw2[i * 2 + 0];
        o1 += h * fcw2[i * 2 + 1];
    }
    out[g * 2 + 0] = o0;
    out[g * 2 + 1] = o1;
}

// ---------------------------------------------------------------------------

extern "C" void kernel_launch(void* const* d_in, const int* in_sizes, int n_in,
                              void* d_out, int out_size, void* d_ws, size_t ws_size,
                              hipStream_t stream) {
    (void)in_sizes; (void)n_in; (void)out_size; (void)ws_size;

    const float* x     = (const float*)d_in[0];
    const int*   eidx  = (const int*)  d_in[1];   // [2, N_EDGES]
    const int*   batch = (const int*)  d_in[2];
    const float* W1    = (const float*)d_in[3];
    const float* b1    = (const float*)d_in[4];
    const float* W2    = (const float*)d_in[5];
    const float* b2    = (const float*)d_in[6];
    const float* fcw1  = (const float*)d_in[7];
    const float* fcb1  = (const float*)d_in[8];
    const float* fcw2  = (const float*)d_in[9];
    const float* fcb2  = (const float*)d_in[10];
    float* out = (float*)d_out;

    const int* esrc = eidx;
    const int* edst = eidx + N_EDGES;

    // workspace layout (~85 MB, all pow2-aligned offsets)
    char* ws = (char*)d_ws;
    int*   deg    = (int*)  (ws);                                 // 512 KB
    float* dinv   = (float*)(ws + (size_t)512 * 1024);            // 512 KB
    float* h1     = (float*)(ws + (size_t)1   * 1024 * 1024);     // 8 MB
    float* agg1   = (float*)(ws + (size_t)9   * 1024 * 1024);     // 8 MB (becomes h1post)
    float* hW     = (float*)(ws + (size_t)17  * 1024 * 1024);     // 32 MB
    float* agg2   = (float*)(ws + (size_t)49  * 1024 * 1024);     // 32 MB
    float* pooled = (float*)(ws + (size_t)81  * 1024 * 1024);     // 32 KB

    const int B = 256;

    k_init_deg   <<<N_NODES / B, B, 0, stream>>>(deg);
    k_degree     <<<N_EDGES / B, B, 0, stream>>>(edst, deg);
    k_dinv       <<<N_NODES / B, B, 0, stream>>>(deg, dinv);

    k_conv1_node <<<N_NODES / B, B, 0, stream>>>(x, W1, dinv, h1, agg1);
    k_scatter16  <<<(N_EDGES * 4) / B, B, 0, stream>>>(esrc, edst, dinv, h1, agg1);
    k_bias_relu16<<<(N_NODES * 16) / B, B, 0, stream>>>(agg1, b1);

    // 8192 row-tiles * 4 col-tiles = 32768 wave-tiles, 8 waves per block
    k_gemm16_64_wmma<<<32768 / 8, B, 0, stream>>>(agg1, W2, dinv, hW, agg2);

    k_scatter64  <<<(N_EDGES * 16) / B, B, 0, stream>>>(esrc, edst, dinv, hW, agg2);

    k_pool       <<<NUM_GRAPHS, B, 0, stream>>>(agg2, b2, batch, pooled);
    k_mlp        <<<1, 128, 0, stream>>>(pooled, fcw1, fcb1, fcw2, fcb2, out);
}